// DotGAT_49606872269209
// MI455X (gfx1250) — compile-verified
//
#include <hip/hip_runtime.h>
#include <math.h>

#define NN 50000
#define EE 500000
#define HH 4
#define DD 64
#define HD 256   // H*D

typedef __attribute__((ext_vector_type(2))) float v2f;
typedef __attribute__((ext_vector_type(8))) float v8f;

// ---------------------------------------------------------------------------
// GEMM: Y[nrows x 256] = X[nrows x K] @ W[K x 256], fp32 WMMA 16x16x4.
// Block = 128 threads (4 waves). Block computes a 16x256 output strip:
//   - 16 x K A-tile (contiguous rows) staged into LDS once per block via
//     GLOBAL_LOAD_ASYNC_TO_LDS_B128 (ASYNCcnt), shared by all 4 waves.
//   - each wave computes 16x64 (4 C accumulators), reusing the A fragment
//     read from LDS (ds_load_b64) across 4 WMMAs per k-step.
// nrows % 16 == 0, K % 4 == 0, ncols == 256. EXEC all-ones (no divergence).
// ---------------------------------------------------------------------------
__global__ __launch_bounds__(128)
void gemm16x256(const float* __restrict__ X, const float* __restrict__ W,
                float* __restrict__ Y, int K, int ncols) {
  __shared__ float Atile[16 * 256];            // max K = 256 -> 16 KB of 320 KB
  const int tid  = threadIdx.x;
  const int wave = tid >> 5;
  const int lane = tid & 31;
  const int row0 = blockIdx.x * 16;

  // ---- async-stage A tile: rows row0..row0+15 are one contiguous span ----
  {
    const float* gsrc = X + (size_t)row0 * K;  // 16*K contiguous floats
    const int total4 = (16 * K) >> 2;          // number of 16-byte packets
    for (int i = tid; i < total4; i += 128) {  // uniform trip count, no divergence
      unsigned lds_off = (unsigned)(size_t)(&Atile[i << 2]);
      unsigned long long ga = (unsigned long long)(gsrc + (i << 2));
      asm volatile("global_load_async_to_lds_b128 %0, %1, off"
                   :: "v"(lds_off), "v"(ga) : "memory");
    }
    asm volatile("s_wait_asynccnt 0x0" ::: "memory");
    __syncthreads();
  }

  // ---- register-blocked WMMA compute ----
  const int m    = lane & 15;                  // A row / B,C col within tile
  const int koff = (lane >> 4) << 1;           // lanes 0-15 -> K{0,1}; 16-31 -> K{2,3}
  const float2* arow = (const float2*)(Atile + m * K);   // 8B-aligned (K even)
  const int colBase = wave * 64;
  const float* b0 = W + colBase +  0 + m;
  const float* b1 = W + colBase + 16 + m;
  const float* b2 = W + colBase + 32 + m;
  const float* b3 = W + colBase + 48 + m;

  v8f c0 = {}, c1 = {}, c2 = {}, c3 = {};
#pragma unroll 4
  for (int k0 = 0; k0 < K; k0 += 4) {
    const int kk = k0 + koff;
    float2 af = arow[kk >> 1];                 // ds_load_b64 from LDS
    v2f a; a.x = af.x; a.y = af.y;
    const size_t rk = (size_t)kk * ncols;
    __builtin_prefetch(b0 + rk + 4 * ncols, 0, 3);   // WGP-scope prefetch on B
    v2f b;
    b.x = b0[rk]; b.y = b0[rk + ncols];
    c0 = __builtin_amdgcn_wmma_f32_16x16x4_f32(false, a, false, b, (short)0, c0, false, false);
    b.x = b1[rk]; b.y = b1[rk + ncols];
    c1 = __builtin_amdgcn_wmma_f32_16x16x4_f32(false, a, false, b, (short)0, c1, false, false);
    b.x = b2[rk]; b.y = b2[rk + ncols];
    c2 = __builtin_amdgcn_wmma_f32_16x16x4_f32(false, a, false, b, (short)0, c2, false, false);
    b.x = b3[rk]; b.y = b3[rk + ncols];
    c3 = __builtin_amdgcn_wmma_f32_16x16x4_f32(false, a, false, b, (short)0, c3, false, false);
  }

  // C/D layout: VGPR i -> lanes 0-15: M=i, lanes 16-31: M=i+8; N = lane&15
  const int r0 = row0 + ((lane >> 4) << 3);
  float* yp = Y + (size_t)r0 * ncols + colBase + m;
#pragma unroll
  for (int i = 0; i < 8; ++i) {
    const size_t o = (size_t)i * ncols;
    yp[o]      = c0[i];
    yp[o + 16] = c1[i];
    yp[o + 32] = c2[i];
    yp[o + 48] = c3[i];
  }
}

// ---------------------------------------------------------------------------
// Utility fill
// ---------------------------------------------------------------------------
__global__ void fill_f32(float* __restrict__ p, int n, float v) {
  int i = blockIdx.x * blockDim.x + threadIdx.x;
  if (i < n) p[i] = v;
}

// float atomic-max via signed/unsigned integer ordering trick
__device__ inline void atomicMaxFloat(float* addr, float val) {
  if (!(val < 0.0f)) {
    atomicMax((int*)addr, __float_as_int(val));
  } else {
    atomicMin((unsigned int*)addr, (unsigned int)__float_as_int(val));
  }
}

// ---------------------------------------------------------------------------
// Edge attention scores: one wave per (edge, head).
// a[e,h] = dot(ft[src,h,:], ft[dst,h,:]) / sqrt(D);  amax[dst,h] = max(...)
// ---------------------------------------------------------------------------
__global__ __launch_bounds__(256)
void edge_scores(const float* __restrict__ ft, const int* __restrict__ src,
                 const int* __restrict__ dst, float* __restrict__ a,
                 float* __restrict__ amax) {
  int wid  = (blockIdx.x * blockDim.x + threadIdx.x) >> 5;
  int lane = threadIdx.x & 31;
  if (wid >= EE * HH) return;                       // whole wave exits together
  int e = wid >> 2, h = wid & 3;
  int s = src[e], d = dst[e];
  const float2* fs = (const float2*)(ft + (size_t)s * HD + h * DD);
  const float2* fd = (const float2*)(ft + (size_t)d * HD + h * DD);
  float2 u = fs[lane], v = fd[lane];
  float sum = u.x * v.x + u.y * v.y;
#pragma unroll
  for (int off = 16; off > 0; off >>= 1) sum += __shfl_down(sum, off, 32);
  if (lane == 0) {
    float val = sum * 0.125f;                       // 1/sqrt(64)
    a[wid] = val;
    atomicMaxFloat(&amax[d * HH + h], val);
  }
}

// ---------------------------------------------------------------------------
// ex = exp(a - amax[dst]);  denom[dst,h] += ex   (one thread per (edge,head))
// ---------------------------------------------------------------------------
__global__ __launch_bounds__(256)
void edge_exp(float* __restrict__ a, const int* __restrict__ dst,
              const float* __restrict__ amax, float* __restrict__ denom) {
  int i = blockIdx.x * blockDim.x + threadIdx.x;
  if (i >= EE * HH) return;
  int e = i >> 2, h = i & 3;
  int d = dst[e];
  float ex = __expf(a[i] - amax[d * HH + h]);
  a[i] = ex;
  atomicAdd(&denom[d * HH + h], ex);
}

// ---------------------------------------------------------------------------
// agg[dst,h,:] += ft[src,h,:] * (ex / denom[dst,h])  (one wave per (edge,head))
// ---------------------------------------------------------------------------
__global__ __launch_bounds__(256)
void edge_agg(const float* __restrict__ ft, const int* __restrict__ src,
              const int* __restrict__ dst, const float* __restrict__ a,
              const float* __restrict__ denom, float* __restrict__ agg) {
  int wid  = (blockIdx.x * blockDim.x + threadIdx.x) >> 5;
  int lane = threadIdx.x & 31;
  if (wid >= EE * HH) return;
  int e = wid >> 2, h = wid & 3;
  int s = src[e], d = dst[e];
  float sa = a[wid] / denom[d * HH + h];
  const float2* fs = (const float2*)(ft + (size_t)s * HD + h * DD);
  float2 u = fs[lane];
  float* out = agg + (size_t)d * HD + h * DD + lane * 2;
  atomicAdd(out,     u.x * sa);
  atomicAdd(out + 1, u.y * sa);
}

// ---------------------------------------------------------------------------
// In-place ReLU
// ---------------------------------------------------------------------------
__global__ __launch_bounds__(256)
void relu_inplace(float* __restrict__ p, int n) {
  int i = blockIdx.x * blockDim.x + threadIdx.x;
  if (i < n) p[i] = fmaxf(p[i], 0.0f);
}

// ---------------------------------------------------------------------------
// Host-side launch sequence
// ---------------------------------------------------------------------------
extern "C" void kernel_launch(void* const* d_in, const int* in_sizes, int n_in,
                              void* d_out, int out_size, void* d_ws, size_t ws_size,
                              hipStream_t stream) {
  const float* x     = (const float*)d_in[0];  // [N,128]
  const int*   src   = (const int*)  d_in[1];  // [E]
  const int*   dst   = (const int*)  d_in[2];  // [E]
  const float* W1    = (const float*)d_in[3];  // [128,256]
  const float* Wres1 = (const float*)d_in[4];  // [128,256]
  const float* W2    = (const float*)d_in[5];  // [256,256]
  float* out = (float*)d_out;                  // [N,256]

  // workspace layout (floats)
  float* bufA  = (float*)d_ws;                 // ft (layer feature), N*HD
  float* bufB  = bufA + (size_t)NN * HD;       // residual/agg -> h1 -> (reused)
  float* aBuf  = bufB + (size_t)NN * HD;       // edge scores / ex, E*H
  float* amax  = aBuf + (size_t)EE * HH;       // N*H
  float* denom = amax + (size_t)NN * HH;       // N*H

  const dim3 gemmGrid(NN / 16, 1);             // block covers full 256-col strip
  const int nEH        = EE * HH;              // 2,000,000
  const int waveBlocks = nEH * 32 / 256;       // one wave per (e,h), 8 waves/block
  const int ehBlocks   = (nEH + 255) / 256;
  const int nhBlocks   = (NN * HH + 255) / 256;
  const int featBlocks = (NN * HD + 255) / 256;
  const float NEG_INF  = -__builtin_huge_valf();

  // ---- Layer 1 ----
  gemm16x256<<<gemmGrid, 128, 0, stream>>>(x, W1, bufA, 128, HD);      // ft1
  gemm16x256<<<gemmGrid, 128, 0, stream>>>(x, Wres1, bufB, 128, HD);   // residual = agg init
  fill_f32<<<nhBlocks, 256, 0, stream>>>(amax, NN * HH, NEG_INF);
  fill_f32<<<nhBlocks, 256, 0, stream>>>(denom, NN * HH, 0.0f);
  edge_scores<<<waveBlocks, 256, 0, stream>>>(bufA, src, dst, aBuf, amax);
  edge_exp<<<ehBlocks, 256, 0, stream>>>(aBuf, dst, amax, denom);
  edge_agg<<<waveBlocks, 256, 0, stream>>>(bufA, src, dst, aBuf, denom, bufB);
  relu_inplace<<<featBlocks, 256, 0, stream>>>(bufB, NN * HD);         // h1 in bufB

  // ---- Layer 2 ----
  gemm16x256<<<gemmGrid, 128, 0, stream>>>(bufB, W2, bufA, HD, HD);    // ft2
  fill_f32<<<nhBlocks, 256, 0, stream>>>(amax, NN * HH, NEG_INF);
  fill_f32<<<nhBlocks, 256, 0, stream>>>(denom, NN * HH, 0.0f);
  edge_scores<<<waveBlocks, 256, 0, stream>>>(bufA, src, dst, aBuf, amax);
  edge_exp<<<ehBlocks, 256, 0, stream>>>(aBuf, dst, amax, denom);
  fill_f32<<<featBlocks, 256, 0, stream>>>(out, NN * HD, 0.0f);        // d_out is poisoned
  edge_agg<<<waveBlocks, 256, 0, stream>>>(bufA, src, dst, aBuf, denom, out);
}